// PatchSampler_17480516895327
// MI455X (gfx1250) — compile-verified
//
#include <hip/hip_runtime.h>
#include <stdint.h>

#define PATCH_N 343   // 7*7*7 patches per batch
#define KSEL    8
#define BATCH   4
#define NITER   11    // ceil(343/32) slots per lane (wave32)

// ---------------------------------------------------------------------------
// Kernel 1: deterministic gumbel-style iterative top-k (softmax / argmax /
// knock-out with -inf). One wave (32 lanes) per batch, 4 waves in one block.
// ---------------------------------------------------------------------------
__global__ void __launch_bounds__(128)
gumbel_topk_kernel(const float* __restrict__ logits,
                   float* __restrict__ soft_out,   // [k, B, N]
                   float* __restrict__ one_out,    // [k, B, N]
                   float* __restrict__ idx_out_f,  // [k, B] as float
                   int*   __restrict__ idx_out_i)  // [k, B] int scratch
{
    const int lane = threadIdx.x & 31;
    const int b    = threadIdx.x >> 5;
    const float tau = (float)(2.0 / 3.0);
    const float NEG_INF = -__builtin_inff();

    // z = logits / tau ; invalid slots = -inf
    float z[NITER];
#pragma unroll
    for (int j = 0; j < NITER; ++j) {
        const int idx = lane + 32 * j;
        z[j] = (idx < PATCH_N) ? (logits[b * PATCH_N + idx] / tau) : NEG_INF;
    }

    for (int t = 0; t < KSEL; ++t) {
        // ---- argmax of z (first occurrence on ties, like jnp.argmax) ----
        float bm = NEG_INF;
        int   bi = PATCH_N;
#pragma unroll
        for (int j = 0; j < NITER; ++j) {
            const int idx = lane + 32 * j;
            if (z[j] > bm) { bm = z[j]; bi = idx; }   // ascending -> lowest idx kept
        }
#pragma unroll
        for (int off = 16; off > 0; off >>= 1) {
            const float om = __shfl_xor(bm, off, 32);
            const int   oi = __shfl_xor(bi, off, 32);
            if (om > bm || (om == bm && oi < bi)) { bm = om; bi = oi; }
        }

        // ---- softmax(z) with max subtraction (exp(-inf - bm) == 0) ----
        float e[NITER];
        float s = 0.0f;
#pragma unroll
        for (int j = 0; j < NITER; ++j) {
            e[j] = expf(z[j] - bm);
            s += e[j];
        }
#pragma unroll
        for (int off = 16; off > 0; off >>= 1) s += __shfl_xor(s, off, 32);

        const int base = (t * BATCH + b) * PATCH_N;
#pragma unroll
        for (int j = 0; j < NITER; ++j) {
            const int idx = lane + 32 * j;
            if (idx < PATCH_N) {
                soft_out[base + idx] = e[j] / s;
                one_out [base + idx] = (idx == bi) ? 1.0f : 0.0f;
            }
            if (idx == bi) z[j] = NEG_INF;   // knock out for next round
        }
        if (lane == 0) {
            idx_out_f[t * BATCH + b] = (float)bi;
            idx_out_i[t * BATCH + b] = bi;
        }
    }
}

// ---------------------------------------------------------------------------
// Kernel 2: patch gather via CDNA5 async global->LDS->global engine.
// Grid: (patch p = t*B+b in 0..31, z-slice in 0..31), 256 threads/block.
// Each thread moves one 16B chunk: global_load_async_to_lds_b128 then
// global_store_async_from_lds_b128 (ASYNCcnt-tracked, no VGPR data path).
// ---------------------------------------------------------------------------
#ifndef USE_ASYNC_LDS
#define USE_ASYNC_LDS 1
#endif

__global__ void __launch_bounds__(256)
gather_kernel(const float* __restrict__ vol,   // [B,1,128,128,128]
              const int*   __restrict__ sel,   // [k*B]
              float*       __restrict__ out)   // [k*B,1,32,32,32]
{
    const int p   = blockIdx.x;        // patch slot (t*BATCH + b)
    const int zz  = blockIdx.y;        // z within patch
    const int b   = p & (BATCH - 1);
    const int idx = sel[p];

    const int nd  = idx / 49;
    const int rem = idx - nd * 49;
    const int nh  = rem / 7;
    const int nw  = rem - nh * 7;

    const int tid = threadIdx.x;
    const int y   = tid >> 3;          // 0..31 row within slice
    const int xi  = tid & 7;           // 0..7  float4 within row

    const size_t srcOff =
        ((((size_t)b * 128 + (size_t)(nd * 16 + zz)) * 128) + (size_t)(nh * 16 + y)) * 128
        + (size_t)(nw * 16) + (size_t)(xi * 4);
    const size_t dstOff = (size_t)p * 32768 + (size_t)zz * 1024 + (size_t)(y * 32 + xi * 4);

#if USE_ASYNC_LDS
    __shared__ float buf[1024];        // 4 KB staging slice
    const uint32_t lds  = (uint32_t)(size_t)&buf[tid * 4];
    const uint64_t gsrc = (uint64_t)(const void*)(vol + srcOff);
    const uint64_t gdst = (uint64_t)(void*)(out + dstOff);

    asm volatile("global_load_async_to_lds_b128 %0, %1, off"
                 :: "v"(lds), "v"(gsrc) : "memory");
    asm volatile("s_wait_asynccnt 0" ::: "memory");
    asm volatile("global_store_async_from_lds_b128 %0, %1, off"
                 :: "v"(gdst), "v"(lds) : "memory");
    asm volatile("s_wait_asynccnt 0" ::: "memory");
#else
    const float4 v = *(const float4*)(vol + srcOff);
    *(float4*)(out + dstOff) = v;
#endif
}

// ---------------------------------------------------------------------------
extern "C" void kernel_launch(void* const* d_in, const int* in_sizes, int n_in,
                              void* d_out, int out_size, void* d_ws, size_t ws_size,
                              hipStream_t stream)
{
    const float* volume = (const float*)d_in[0];   // (4,1,128,128,128) f32
    const float* logits = (const float*)d_in[1];   // (4,1,7,7,7) f32
    // d_in[2] is k == 8 (compile-time constant KSEL)

    float* out         = (float*)d_out;
    float* sel_patches = out;                                    // 1,048,576
    float* soft_hots   = out + 1048576;                          //    10,976
    float* one_hots    = out + 1048576 + 10976;                  //    10,976
    float* idx_f       = out + 1048576 + 2 * 10976;              //        32
    int*   sel_idx     = (int*)d_ws;                             // 32 ints

    hipLaunchKernelGGL(gumbel_topk_kernel, dim3(1), dim3(128), 0, stream,
                       logits, soft_hots, one_hots, idx_f, sel_idx);

    hipLaunchKernelGGL(gather_kernel, dim3(32, 32), dim3(256), 0, stream,
                       volume, sel_idx, sel_patches);
}